// FraudDetectionModel_65481071410092
// MI455X (gfx1250) — compile-verified
//
#include <hip/hip_runtime.h>
#include <stdint.h>

#define DEPTH 4      // async pipeline depth (stages in flight per wave)
#define BLOCK 256    // 8 waves
#define STAGE_B 32   // bytes of x per lane per stage (4 rows * 8B)

typedef float v4f __attribute__((ext_vector_type(4)));  // native vector: OK for nontemporal builtins

// ---------------- fast tanh: prefer hardware v_tanh_f32 on gfx1250 ----------
__device__ __forceinline__ float fast_tanh(float x) {
#if __has_builtin(__builtin_amdgcn_tanhf)
  return __builtin_amdgcn_tanhf(x);
#elif __has_builtin(__builtin_amdgcn_tanh_f32)
  return __builtin_amdgcn_tanh_f32(x);
#else
  // tanh(x) = 1 - 2/(exp(2x)+1); exp(2x) = exp2(x * 2*log2(e))
  float e = __builtin_amdgcn_exp2f(x * 2.885390081777927f);
  return 1.0f - 2.0f * __builtin_amdgcn_rcpf(e + 1.0f);
#endif
}

struct NetP {
  float W[4][4];          // W[l][r*2+c]
  float b[4][2], s[4][2], t[4][2];
  float wf0, wf1, bf;
};

__device__ __forceinline__ float net_row(float h0, float h1, const NetP& p) {
#pragma unroll
  for (int l = 0; l < 4; ++l) {
    float a0 = fmaf(p.W[l][0], h0, fmaf(p.W[l][1], h1, p.b[l][0]));
    float a1 = fmaf(p.W[l][2], h0, fmaf(p.W[l][3], h1, p.b[l][1]));
    h0 = fmaf(p.s[l][0], fast_tanh(a0), p.t[l][0]);
    h1 = fmaf(p.s[l][1], fast_tanh(a1), p.t[l][1]);
  }
  return fmaf(p.wf0, h0, fmaf(p.wf1, h1, p.bf));
}

// Issue one 32B/lane stage: two async b128 global->LDS transfers.
// INST_OFFSET is added to BOTH the LDS and the global address, so one
// lds_addr/gaddr pair covers both halves. s_wait_dscnt 0 guarantees any
// earlier ds_load of this slot fully completed before the DMA overwrites it.
__device__ __forceinline__ void async_stage_load(uint32_t lds_addr, const float* gp) {
  uint64_t ga = (uint64_t)(uintptr_t)gp;
  asm volatile(
      "s_wait_dscnt 0x0\n\t"
      "global_load_async_to_lds_b128 %0, %1, off\n\t"
      "global_load_async_to_lds_b128 %0, %1, off offset:16"
      :: "v"(lds_addr), "v"(ga) : "memory");
}

__global__ void __launch_bounds__(BLOCK)
fraud_mlp_kernel(const float* __restrict__ x,  const float* __restrict__ Ws,
                 const float* __restrict__ bs, const float* __restrict__ sc,
                 const float* __restrict__ sh, const float* __restrict__ Wf,
                 const float* __restrict__ bf, float* __restrict__ out,
                 unsigned n) {
  // Per-wave private staging: DEPTH slots * 32 lanes * 32B. No barriers needed.
  __shared__ __align__(16) unsigned char smem[DEPTH * BLOCK * STAGE_B];

  // Load the 43 network parameters once (uniform addresses -> s_load).
  NetP p;
#pragma unroll
  for (int l = 0; l < 4; ++l) {
#pragma unroll
    for (int k = 0; k < 4; ++k) p.W[l][k] = Ws[l * 4 + k];
#pragma unroll
    for (int k = 0; k < 2; ++k) {
      p.b[l][k] = bs[l * 2 + k];
      p.s[l][k] = sc[l * 2 + k];
      p.t[l][k] = sh[l * 2 + k];
    }
  }
  p.wf0 = Wf[0]; p.wf1 = Wf[1]; p.bf = bf[0];

  // All-32-bit stage indexing (rows <= 2^30 since in_sizes is int).
  const unsigned T     = gridDim.x * BLOCK;
  const unsigned tid   = blockIdx.x * BLOCK + threadIdx.x;
  const unsigned total = (n + 3u) >> 2;  // all stages (incl. partial last)
  const unsigned full  = n >> 2;         // stages with 4 fully-in-range rows

  const int lane = threadIdx.x & 31;
  const int wave = threadIdx.x >> 5;
  const uint32_t lds_lane = (uint32_t)(uintptr_t)smem +
                            (uint32_t)(wave * (DEPTH * 1024) + lane * STAGE_B);

  // -------- prologue: fill the pipeline --------
  int inflight = 0;  // outstanding async instructions this lane accounted for
#pragma unroll
  for (unsigned j = 0; j < DEPTH; ++j) {
    unsigned sj = tid + j * T;
    if (sj < full) {
      async_stage_load(lds_lane + j * 1024u, x + (size_t)sj * 8u);
      inflight += 2;
    }
  }

  // -------- steady state --------
  unsigned s = tid;
  for (unsigned j = 0; s < total; ++j, s += T) {
    const unsigned slot = j & (DEPTH - 1u);
    if (s < full) {
      // Wait until this stage's two async ops (oldest in flight) are done.
      // Per-lane inflight can only undercount the wave counter -> stricter wait.
      if (inflight >= 8)      asm volatile("s_wait_asynccnt 0x6" ::: "memory");
      else if (inflight == 6) asm volatile("s_wait_asynccnt 0x4" ::: "memory");
      else if (inflight == 4) asm volatile("s_wait_asynccnt 0x2" ::: "memory");
      else                    asm volatile("s_wait_asynccnt 0x0" ::: "memory");
      inflight -= 2;

      const v4f* lp = reinterpret_cast<const v4f*>(
          smem + (size_t)(wave * (DEPTH * 1024) + slot * 1024u + lane * STAGE_B));
      v4f v0 = lp[0];
      v4f v1 = lp[1];

      v4f r;
      r.x = net_row(v0.x, v0.y, p);
      r.y = net_row(v0.z, v0.w, p);
      r.z = net_row(v1.x, v1.y, p);
      r.w = net_row(v1.z, v1.w, p);
      // Output is write-once: non-temporal b128 store keeps the 67MB write
      // stream from evicting the 134MB read stream out of the 192MB L2.
      __builtin_nontemporal_store(r, reinterpret_cast<v4f*>(out + (size_t)s * 4u));

      // Refill this slot with the stage DEPTH iterations ahead.
      unsigned s2 = s + DEPTH * T;
      if (s2 < full) {
        async_stage_load(lds_lane + slot * 1024u, x + (size_t)s2 * 8u);
        inflight += 2;
      }
    } else {
      // Partial final stage: direct guarded path (never staged via LDS).
      unsigned base = s * 4u;
      for (unsigned rr = 0; rr < 4; ++rr) {
        unsigned row = base + rr;
        if (row < n) out[row] = net_row(x[(size_t)row * 2],
                                        x[(size_t)row * 2 + 1], p);
      }
    }
  }
}

extern "C" void kernel_launch(void* const* d_in, const int* in_sizes, int n_in,
                              void* d_out, int out_size, void* d_ws, size_t ws_size,
                              hipStream_t stream) {
  (void)n_in; (void)out_size; (void)d_ws; (void)ws_size;
  const float* x  = (const float*)d_in[0];
  const float* Ws = (const float*)d_in[1];
  const float* bs = (const float*)d_in[2];
  const float* sc = (const float*)d_in[3];
  const float* sh = (const float*)d_in[4];
  const float* Wf = (const float*)d_in[5];
  const float* bf = (const float*)d_in[6];
  float* out = (float*)d_out;

  const unsigned n = (unsigned)(in_sizes[0] / 2);        // rows
  const unsigned total_stages = (n + 3u) / 4u;           // 4 rows/thread/stage
  unsigned want_threads = (total_stages + 15u) / 16u;    // ~16 stages/thread
  int grid = (int)((want_threads + BLOCK - 1u) / BLOCK);
  if (grid < 1) grid = 1;

  fraud_mlp_kernel<<<grid, BLOCK, 0, stream>>>(x, Ws, bs, sc, sh, Wf, bf, out, n);
}